// Channel_Embedding_ablation_46703474377299
// MI455X (gfx1250) — compile-verified
//
#include <hip/hip_runtime.h>
#include <hip/hip_bf16.h>

typedef __attribute__((ext_vector_type(2))) float        v2f;
typedef __attribute__((ext_vector_type(8))) float        v8f;
typedef __attribute__((ext_vector_type(4))) unsigned int v4u;
typedef __attribute__((ext_vector_type(8))) int          v8i;
typedef __attribute__((ext_vector_type(4))) int          v4i;

#define B_  32
#define D_  64
#define L_  4096
#define OC_ 32
#define E_  8
#define LP_ 4094      // L - 2 (VALID conv, k=3)
#define TN  64        // output positions per tile
#define XS  66        // LDS row stride for x tile (TDM packs rows contiguously)
#define HS  68        // LDS row stride for h tile
#define WS  33        // LDS row stride for W_eff

// ---------------------------------------------------------------------------
// Kernel 1: gates (softmax + top-2 renorm) folded into conv2 ->
//           W_eff[b][32][32] and b_eff[b][32] in workspace.
// ---------------------------------------------------------------------------
__global__ __launch_bounds__(256) void gates_fold_kernel(
    const float* __restrict__ x, const float* __restrict__ w_gate,
    const float* __restrict__ conv2_w, const float* __restrict__ conv2_b,
    float* __restrict__ weff, float* __restrict__ beff)
{
  __shared__ float gsh[B_ * E_];
  const int tid = threadIdx.x;

  if (tid < B_) {
    const int b = tid;
    float lg[E_];
    #pragma unroll
    for (int e = 0; e < E_; ++e) lg[e] = 0.0f;
    for (int d = 0; d < D_; ++d) {
      const float* xp = x + ((size_t)b * D_ + d) * L_ + (L_ - 6);
      #pragma unroll
      for (int p = 0; p < 5; ++p) {
        const float xv = xp[p];
        const float* wg = w_gate + (d * 5 + p) * E_;
        #pragma unroll
        for (int e = 0; e < E_; ++e) lg[e] += xv * wg[e];
      }
    }
    float mx = lg[0];
    #pragma unroll
    for (int e = 1; e < E_; ++e) mx = fmaxf(mx, lg[e]);
    float s = 0.0f;
    #pragma unroll
    for (int e = 0; e < E_; ++e) { lg[e] = expf(lg[e] - mx); s += lg[e]; }
    const float invs = 1.0f / s;
    #pragma unroll
    for (int e = 0; e < E_; ++e) lg[e] *= invs;
    int i0 = 0;
    #pragma unroll
    for (int e = 1; e < E_; ++e) if (lg[e] > lg[i0]) i0 = e;
    int i1 = (i0 == 0) ? 1 : 0;
    #pragma unroll
    for (int e = 0; e < E_; ++e) if (e != i0 && lg[e] > lg[i1]) i1 = e;
    const float v0 = lg[i0], v1 = lg[i1];
    const float inv = 1.0f / (v0 + v1 + 1e-6f);
    #pragma unroll
    for (int e = 0; e < E_; ++e) gsh[b * E_ + e] = 0.0f;
    gsh[b * E_ + i0] = v0 * inv;
    gsh[b * E_ + i1] = v1 * inv;
  }
  __syncthreads();

  for (int idx = tid; idx < B_ * OC_ * OC_; idx += 256) {
    const int c = idx & 31;
    const int d = (idx >> 5) & 31;
    const int b = idx >> 10;
    float acc = 0.0f;
    #pragma unroll
    for (int e = 0; e < E_; ++e)
      acc += gsh[b * E_ + e] * conv2_w[((d << 3) + e) * OC_ + c];
    weff[idx] = acc;
  }
  for (int idx = tid; idx < B_ * OC_; idx += 256) {
    const int d = idx & 31;
    const int b = idx >> 5;
    float acc = 0.0f;
    #pragma unroll
    for (int e = 0; e < E_; ++e)
      acc += gsh[b * E_ + e] * conv2_b[(d << 3) + e];
    beff[idx] = acc;
  }
}

// ---------------------------------------------------------------------------
// Kernel 2: TDM-staged x tile + conv1(tanh) + gated 32x32 mix on
//           V_WMMA_F32_16X16X4_F32. Grid: (tiles=64, batch=32), 8 wave32.
// ---------------------------------------------------------------------------
__global__ __launch_bounds__(256) void moe_conv_wmma_kernel(
    const float* __restrict__ x, const float* __restrict__ conv1_w,
    const float* __restrict__ conv1_b, const float* __restrict__ weff,
    const float* __restrict__ beff, float* __restrict__ out)
{
  __shared__ float xs[D_ * XS];       // x tile [64][66], TDM-packed
  __shared__ float w1s[OC_ * 192];    // conv1 weights [32][192] (c*3+t)
  __shared__ float hs[OC_ * HS];      // tanh(conv1) tile [32][68]
  __shared__ float wes[OC_ * WS];     // W_eff[b] [32][33]
  __shared__ float b1s[OC_];

  const int tid  = threadIdx.x;
  const int b    = blockIdx.y;
  const int tile = blockIdx.x;
  const int l0   = tile * TN;

  // ---- Tensor Data Mover: DMA the 64x66 fp32 x-tile into LDS (wave 0) ----
  if (tid < 32) {
    const unsigned lds_base = (unsigned)(size_t)(&xs[0]);
    const unsigned long long ga =
        (unsigned long long)(size_t)(x + ((size_t)b * D_) * L_ + l0);
    // D# group 0: count=1 | lds_addr | global_addr[56:0] | type=2
    v4u g0;
    g0.x = 1u;                                   // count=1 (valid), user mode
    g0.y = lds_base;                             // lds_addr
    g0.z = (unsigned)ga;                         // global_addr[31:0]
    g0.w = ((unsigned)(ga >> 32) & 0x01FFFFFFu)  // global_addr[56:32]
           | (2u << 30);                         // type=2 ("image")
    // D# group 1: data_size=4B, tensor 4096x64, tile 66x64, stride 4096
    v8i g1;
    g1[0] = (int)(2u << 16);          // data_size code 2 (=4 bytes)
    g1[1] = (int)(4096u << 16);       // tensor_dim0[15:0] in [31:16]
    g1[2] = (int)(64u << 16);         // tensor_dim0 hi=0 | tensor_dim1[15:0]
    g1[3] = (int)(66u << 16);         // tensor_dim1 hi=0 | tile_dim0=66
    g1[4] = 64;                       // tile_dim1=64 | tile_dim2=0
    g1[5] = 4096;                     // tensor_dim0_stride[31:0]
    g1[6] = 0;                        // stride0 hi | stride1 lo
    g1[7] = 0;                        // stride1 hi
    v4i gz = {0, 0, 0, 0};
#if __clang_major__ >= 23
    v8i gz8 = {0, 0, 0, 0, 0, 0, 0, 0};
    __builtin_amdgcn_tensor_load_to_lds(g0, g1, gz, gz, gz8, 0);
#else
    __builtin_amdgcn_tensor_load_to_lds(g0, g1, gz, gz, 0);
#endif
  }

  // ---- other waves stage weights while the TDM streams x ----
  for (int idx = tid; idx < OC_ * 192; idx += 256) w1s[idx] = conv1_w[idx];
  for (int idx = tid; idx < OC_ * OC_; idx += 256) {
    const int d = idx >> 5, c = idx & 31;
    wes[d * WS + c] = weff[((size_t)b << 10) + idx];
  }
  if (tid < OC_) b1s[tid] = conv1_b[tid];

  // prefetch next tile's x into L2 while we compute this one
  if (tile + 1 < gridDim.x) {
    const float* nx = x + ((size_t)b * D_ + (tid >> 2)) * L_ + (l0 + TN) + ((tid & 3) << 4);
    __builtin_prefetch(nx, 0, 1);
  }

  if (tid < 32) __builtin_amdgcn_s_wait_tensorcnt(0);
  __syncthreads();

  const int wave = tid >> 5;
  const int lane = tid & 31;
  const int mt   = wave & 1;      // M-tile: rows [0..15] or [16..31]
  const int nt   = wave >> 1;     // N-tile: 16-col slice of the 64-wide tile
  const int lrow = lane & 15;
  const int lhi  = lane >> 4;     // K half-select per ISA 16x4 fp32 A layout

  // ---- stage 1: conv1 as GEMM, tap-major (K = channels, 4 per WMMA) ----
  v8f acc = {};
  #pragma unroll
  for (int t = 0; t < 3; ++t) {
    #pragma unroll
    for (int kk = 0; kk < 16; ++kk) {
      const int c0 = kk * 4;
      v2f a, bf;
      const int ka = c0 + (lhi << 1);                    // A: K = ka, ka+1
      a.x = w1s[(mt * 16 + lrow) * 192 + (ka + 0) * 3 + t];
      a.y = w1s[(mt * 16 + lrow) * 192 + (ka + 1) * 3 + t];
      const int kb = c0 + lhi;                           // B: rows kb, kb+2
      bf.x = xs[(kb + 0) * XS + nt * 16 + lrow + t];
      bf.y = xs[(kb + 2) * XS + nt * 16 + lrow + t];
      acc = __builtin_amdgcn_wmma_f32_16x16x4_f32(false, a, false, bf,
                                                  (short)0, acc, false, false);
    }
  }

  // bias + tanh -> hs  (C/D layout: vgpr r, lanes0-15 M=r, lanes16-31 M=r+8)
  #pragma unroll
  for (int r = 0; r < 8; ++r) {
    const int oc = mt * 16 + r + (lhi << 3);
    const int n  = nt * 16 + lrow;
    hs[oc * HS + n] = tanhf(acc[r] + b1s[oc]);
  }
  __syncthreads();

  // ---- stage 2: gated expert mix == per-batch 32x32 GEMM on WMMA ----
  v8f mo = {};
  #pragma unroll
  for (int kk = 0; kk < 8; ++kk) {
    const int k0 = kk * 4;
    v2f a, bf;
    const int ka = k0 + (lhi << 1);
    a.x = wes[(mt * 16 + lrow) * WS + ka + 0];
    a.y = wes[(mt * 16 + lrow) * WS + ka + 1];
    const int kb = k0 + lhi;
    bf.x = hs[(kb + 0) * HS + nt * 16 + lrow];
    bf.y = hs[(kb + 2) * HS + nt * 16 + lrow];
    mo = __builtin_amdgcn_wmma_f32_16x16x4_f32(false, a, false, bf,
                                               (short)0, mo, false, false);
  }

  #pragma unroll
  for (int r = 0; r < 8; ++r) {
    const int d = mt * 16 + r + (lhi << 3);
    const int n = nt * 16 + lrow;
    const int l = l0 + n;
    if (l < LP_)
      out[((size_t)b * OC_ + d) * LP_ + l] = mo[r] + beff[b * OC_ + d];
  }
}

// ---------------------------------------------------------------------------
extern "C" void kernel_launch(void* const* d_in, const int* in_sizes, int n_in,
                              void* d_out, int out_size, void* d_ws, size_t ws_size,
                              hipStream_t stream) {
  const float* x       = (const float*)d_in[0];
  const float* w_gate  = (const float*)d_in[1];
  const float* conv1_w = (const float*)d_in[2];
  const float* conv1_b = (const float*)d_in[3];
  const float* conv2_w = (const float*)d_in[4];
  const float* conv2_b = (const float*)d_in[5];
  float* out = (float*)d_out;

  float* weff = (float*)d_ws;                    // 32*32*32 floats
  float* beff = weff + B_ * OC_ * OC_;           // 32*32 floats

  gates_fold_kernel<<<1, 256, 0, stream>>>(x, w_gate, conv2_w, conv2_b, weff, beff);

  dim3 grid((LP_ + TN - 1) / TN, B_);            // (64, 32)
  moe_conv_wmma_kernel<<<grid, 256, 0, stream>>>(x, conv1_w, conv1_b,
                                                 weff, beff, out);
}